// Feature_80745385165455
// MI455X (gfx1250) — compile-verified
//
#include <hip/hip_runtime.h>
#include <math.h>

#define BATCH 16
#define NATOM 96
#define KRAD  16
#define KANG  8
#define NFEAT 24
#define NPAIRS 4560               // C(96,2)
#define PI_F 3.14159265358979323846f
#define RCR_F 5.2f
#define RCA_F 3.5f
#define LOG2E_F 1.4426950408889634f

typedef float v2f __attribute__((ext_vector_type(2)));
typedef float v8f __attribute__((ext_vector_type(8)));
typedef unsigned int u32x4 __attribute__((ext_vector_type(4)));
typedef unsigned int u32x8 __attribute__((ext_vector_type(8)));

// ---------------------------------------------------------------------------
// Kernel 1: per batch, G = X*X^T via V_WMMA_F32_16X16X4_F32 (K padded 3->4),
// fused into D2[i][j] = |xi|^2 + |xj|^2 - 2*Gij.  6x6 tile grid, 9 waves x 4.
// ---------------------------------------------------------------------------
__global__ __launch_bounds__(288) void gram_d2_kernel(const float* __restrict__ coords,
                                                      float* __restrict__ d2) {
    __shared__ float Xs[NATOM * 4];   // padded coords
    __shared__ float n2[NATOM];       // squared norms
    const int b   = blockIdx.x;
    const int tid = threadIdx.x;

    for (int idx = tid; idx < NATOM * 4; idx += 288) {
        int m = idx >> 2, c = idx & 3;
        Xs[idx] = (c < 3) ? coords[(b * NATOM + m) * 3 + c] : 0.0f;
    }
    __syncthreads();
    for (int m = tid; m < NATOM; m += 288) {
        float x = Xs[m * 4 + 0], y = Xs[m * 4 + 1], z = Xs[m * 4 + 2];
        n2[m] = x * x + y * y + z * z;
    }
    __syncthreads();

    const int wave = tid >> 5;     // 0..8
    const int lane = tid & 31;
    const int half = lane >> 4;    // 0: K=0/1, 1: K=2/3
    const int l16  = lane & 15;

    #pragma unroll
    for (int q = 0; q < 4; ++q) {
        const int t  = wave * 4 + q;       // tile 0..35
        const int ti = t / 6, tj = t % 6;

        // A: 16x4, lane -> M=l16, VGPR0 = K=(half?2:0), VGPR1 = K=(half?3:1)
        const int ra = (ti * 16 + l16) * 4;
        const int rb = (tj * 16 + l16) * 4;
        v2f a, bm;
        a.x  = Xs[ra + (half ? 2 : 0)];
        a.y  = Xs[ra + (half ? 3 : 1)];
        bm.x = Xs[rb + (half ? 2 : 0)];    // B = Xj^T, same per-lane layout
        bm.y = Xs[rb + (half ? 3 : 1)];

        v8f c = {};
        c = __builtin_amdgcn_wmma_f32_16x16x4_f32(
                /*neg_a=*/false, a, /*neg_b=*/false, bm,
                /*c_mod=*/(short)0, c, /*reuse_a=*/false, /*reuse_b=*/false);

        #pragma unroll
        for (int r = 0; r < 8; ++r) {
            const int m = ti * 16 + r + half * 8;   // C/D layout: M=r (+8 for hi half)
            const int n = tj * 16 + l16;
            float v = n2[m] + n2[n] - 2.0f * c[r];
            d2[b * (NATOM * NATOM) + m * NATOM + n] = fmaxf(v, 0.0f);
        }
    }
}

// ---------------------------------------------------------------------------
// Kernel 2: one block per (atom i, batch b).  Full-batch D2 matrix (36 KB)
// DMA'd into LDS by the Tensor Data Mover (one tensor_load_to_lds per block,
// TENSORcnt-tracked), then:
//   radial:  g2[k] = sum_j exp(-er_k (d_ij - sr_k)^2) * fcR(d_ij)
//   angular: g3[k] = sum_{j<k} 2^(1-ze) (1+cos)^ze exp(-ea*s) * fA12 fA13 fA23
// One exp2 per (pair,k):  exp2((1-ze) + ze*log2(1+cos) - ea*log2e*s) * F
// Deterministic reductions (fixed-order tree, no float atomics).
// ---------------------------------------------------------------------------
__global__ __launch_bounds__(256) void acsf_kernel(const float* __restrict__ d2g,
                                                   const int*   __restrict__ atom_types,
                                                   const float* __restrict__ EtaR,
                                                   const float* __restrict__ ShfR,
                                                   const float* __restrict__ Zeta,
                                                   const float* __restrict__ EtaA,
                                                   float* __restrict__ out) {
    __shared__ float D2s[NATOM * NATOM];               // 36 KB
    __shared__ float di[NATOM], fAi[NATOM], fRi[NATOM];
    __shared__ float erS[KRAD], srS[KRAD];
    __shared__ float zeS[KANG], c0S[KANG], ea2S[KANG];
    __shared__ float red[256 * KANG];                  // 8 KB
    __shared__ float g2S[KRAD];

    const int i   = blockIdx.x;
    const int b   = blockIdx.y;
    const int tid = threadIdx.x;

    // --- TDM: DMA the batch's 96x96 f32 D2 tile from global into LDS --------
    if (tid < 32) {   // wave 0 only (branchy skip for waves 1..7)
        const unsigned long long gaddr =
            (unsigned long long)(uintptr_t)(d2g + (size_t)b * NATOM * NATOM);
        const unsigned ldsa = (unsigned)(uintptr_t)(&D2s[0]);   // LDS byte offset

        // D# group 0: count=1 | lds_addr | global_addr[56:0] | type=2
        u32x4 g0;
        g0.x = 1u;                                   // count=1, user mode
        g0.y = ldsa;                                 // lds_addr (bytes)
        g0.z = (unsigned)(gaddr & 0xFFFFFFFFull);    // global_addr[31:0]
        g0.w = (unsigned)((gaddr >> 32) & 0x01FFFFFFu) | (2u << 30);  // [56:32]|type=2

        // D# group 1: data_size=4B, tensor 96x96, tile 96x96, stride0=96
        u32x8 g1;
        g1.s0 = 2u << 16;                            // workgroup_mask=0, data_size=2(4B)
        g1.s1 = (unsigned)NATOM << 16;               // tensor_dim0[15:0] @ bits[63:48]
        g1.s2 = (unsigned)NATOM << 16;               // tensor_dim1[15:0] @ bits[95:80]
        g1.s3 = (unsigned)NATOM << 16;               // tile_dim0        @ bits[127:112]
        g1.s4 = (unsigned)NATOM;                     // tile_dim1        @ bits[143:128]
        g1.s5 = (unsigned)NATOM;                     // tensor_dim0_stride[31:0]
        g1.s6 = 0u;
        g1.s7 = 0u;

        asm volatile("tensor_load_to_lds %0, %1"
                     :: "s"(g0), "s"(g1)
                     : "memory");
        __builtin_amdgcn_s_wait_tensorcnt(0);        // TDM done before barrier
    }

    const int ty = atom_types[i];
    if (tid >= 64 && tid < 64 + KRAD) {
        int q = tid - 64;
        erS[q] = EtaR[ty * KRAD + q];
        srS[q] = ShfR[ty * KRAD + q];
    } else if (tid >= 96 && tid < 96 + KANG) {
        int q = tid - 96;
        float ze = Zeta[ty * KANG + q];
        zeS[q]  = ze;
        c0S[q]  = 1.0f - ze;                         // 2^(1-ze) folded into exp2 arg
        ea2S[q] = EtaA[ty * KANG + q] * LOG2E_F;     // exp(-ea*s) -> exp2(-ea*log2e*s)
    }
    __syncthreads();

    // distances + cutoff factors for row i
    for (int j = tid; j < NATOM; j += 256) {
        float d = sqrtf(D2s[i * NATOM + j]);
        di[j]  = d;
        fRi[j] = 0.5f * cosf(d * (PI_F / RCR_F)) + 0.5f;
        fAi[j] = 0.5f * cosf(d * (PI_F / RCA_F)) + 0.5f;
    }
    __syncthreads();

    // radial: one thread per k, fixed summation order -> deterministic
    if (tid < KRAD) {
        float er = erS[tid], sr = srS[tid], acc = 0.0f;
        for (int j = 0; j < NATOM; ++j) {
            if (j == i) continue;
            float dd = di[j] - sr;
            acc += expf(-er * dd * dd) * fRi[j];
        }
        g2S[tid] = acc;
    }

    // angular: iterate all unordered pairs {j,k} of 0..95, skip those with i
    float acc[KANG];
    #pragma unroll
    for (int q = 0; q < KANG; ++q) acc[q] = 0.0f;

    for (int p = tid; p < NPAIRS; p += 256) {
        // unrank p -> (j,k), j<k; off(j) = j*(191-j)/2
        int j = (int)((191.0f - sqrtf(36481.0f - 8.0f * (float)p)) * 0.5f);
        if (j < 0) j = 0;
        while ((j + 1) * (190 - j) / 2 <= p) ++j;
        while (j * (191 - j) / 2 > p) --j;
        const int k = j + 1 + (p - j * (191 - j) / 2);
        if (j == i || k == i) continue;

        const float d12  = di[j], d13 = di[k];
        const float d12s = d12 * d12, d13s = d13 * d13;
        const float d23s = D2s[j * NATOM + k];
        const float d23  = sqrtf(d23s);
        const float cosn = (d12s + d13s - d23s) / (2.0f * d12 * d13); // law of cosines
        const float tpl  = fmaxf(1.0f + cosn, 0.0f);
        const float lt   = log2f(tpl);                 // -inf at 0 -> exp2 -> 0 (ok)
        const float s    = d12s + d13s + d23s;
        const float f23  = 0.5f * cosf(d23 * (PI_F / RCA_F)) + 0.5f;
        const float F    = fAi[j] * fAi[k] * f23;

        #pragma unroll
        for (int q = 0; q < KANG; ++q) {
            float arg = c0S[q] + zeS[q] * lt - ea2S[q] * s;
            acc[q] += exp2f(arg) * F;
        }
    }

    #pragma unroll
    for (int q = 0; q < KANG; ++q) red[tid * KANG + q] = acc[q];
    __syncthreads();

    // fixed-order pairwise tree reduction (deterministic)
    for (int sft = 128; sft > 0; sft >>= 1) {
        if (tid < sft) {
            #pragma unroll
            for (int q = 0; q < KANG; ++q)
                red[tid * KANG + q] += red[(tid + sft) * KANG + q];
        }
        __syncthreads();
    }

    float* o = out + ((size_t)b * NATOM + i) * NFEAT;
    if (tid < KRAD) o[tid] = g2S[tid];
    if (tid < KANG) o[KRAD + tid] = red[tid];
}

// ---------------------------------------------------------------------------
extern "C" void kernel_launch(void* const* d_in, const int* in_sizes, int n_in,
                              void* d_out, int out_size, void* d_ws, size_t ws_size,
                              hipStream_t stream) {
    const float* coords     = (const float*)d_in[0];
    const int*   atom_types = (const int*)  d_in[1];
    const float* EtaR       = (const float*)d_in[2];
    const float* ShfR       = (const float*)d_in[3];
    const float* Zeta       = (const float*)d_in[4];
    const float* EtaA       = (const float*)d_in[5];
    float* out = (float*)d_out;
    float* d2  = (float*)d_ws;   // 16*96*96 floats = 576 KB of scratch

    gram_d2_kernel<<<dim3(BATCH), dim3(288), 0, stream>>>(coords, d2);
    acsf_kernel<<<dim3(NATOM, BATCH), dim3(256), 0, stream>>>(
        d2, atom_types, EtaR, ShfR, Zeta, EtaA, out);
}